// STTransformerBlock_21474836480554
// MI455X (gfx1250) — compile-verified
//
#include <hip/hip_runtime.h>
#include <hip/hip_bf16.h>
#include <math.h>

// ---- problem constants (from reference) ----
#define N_NODES 2048
#define R_REG   256
#define T_SEQ   12
#define C_DIM   64
#define H_HEADS 4
#define HD_DIM  16
#define TIME_NUMC 288

typedef __attribute__((ext_vector_type(2))) float v2f;
typedef __attribute__((ext_vector_type(8))) float v8f;

#if defined(__AMDGCN__) && __has_builtin(__builtin_amdgcn_wmma_f32_16x16x4_f32)
#define HAVE_WMMA_F32 1
#endif

// One 16x16x4 f32 WMMA step. Chaining 4 of these gives a 16x16x16 GEMM.
__device__ __forceinline__ v8f wmma_k4(v2f a, v2f b, v8f c) {
#if defined(HAVE_WMMA_F32)
    return __builtin_amdgcn_wmma_f32_16x16x4_f32(
        /*neg_a=*/false, a, /*neg_b=*/false, b,
        /*c_mod=*/(short)0, c, /*reuse_a=*/false, /*reuse_b=*/false);
#else
    (void)a; (void)b;
    return c; // host pass placeholder (never executed)
#endif
}

// Issue an async copy of one 1KB (16x16 f32) tile into LDS: 2 x B128 x 32 lanes.
__device__ __forceinline__ void async_tile_1k(const float* gtile, unsigned ldsv, int lane) {
#if defined(__AMDGCN__)
    unsigned long long ga = (unsigned long long)(size_t)gtile +
                            (unsigned long long)(lane * 16);
    asm volatile("global_load_async_to_lds_b128 %0, %1, off\n\t"
                 "global_load_async_to_lds_b128 %0, %1, off offset:512"
                 :: "v"(ldsv), "v"(ga) : "memory");
#else
    (void)gtile; (void)ldsv; (void)lane;
#endif
}

// =====================================================================
// Elementwise: qa = query + D_S, ka = key_in + D_S ; D_S[n,c]=DS[n]*embW[c]+embb[c]
// =====================================================================
__global__ void k_add_ds(const float* __restrict__ q, const float* __restrict__ kin,
                         const float* __restrict__ DS, const float* __restrict__ embW,
                         const float* __restrict__ embb,
                         float* __restrict__ qa, float* __restrict__ ka, int total) {
    int i = blockIdx.x * blockDim.x + threadIdx.x;
    if (i >= total) return;
    int c = i % C_DIM;
    int n = i / (T_SEQ * C_DIM);
    float ds = DS[n] * embW[c] + embb[c];
    qa[i] = q[i] + ds;
    ka[i] = kin[i] + ds;
}

// =====================================================================
// Per-head projection with PACKED output tiles (layout chosen at compile time).
//   TP==0: Xp[((tt*H + h)*nnodes + node)*16 + e]   (spatial attention)
//   TP==1: Xp[((node*H + h)*16 + tt)*16 + e]       (temporal attention; T=12 of
//          16 tile rows written, rest masked by consumer)
// One wave per (16-token tile, head). 4 chained WMMAs (K=16).
// =====================================================================
template <int TP>
__global__ void k_proj_pack(const float* __restrict__ X, const float* __restrict__ W,
                            float* __restrict__ Xp, int ntok, int nnodes) {
    int lane = threadIdx.x, half = lane >> 4, n16 = lane & 15;
    int tbase = blockIdx.x * 16;
    int h = blockIdx.y;
    if (tbase >= ntok) return;
    const float* Xb = X + (size_t)tbase * C_DIM + h * HD_DIM;
    v8f acc = {0,0,0,0,0,0,0,0};
#pragma unroll
    for (int c = 0; c < 4; ++c) {
        int k0 = c * 4 + half * 2;
        v2f a, b;
        a.x = Xb[(size_t)n16 * C_DIM + k0];
        a.y = Xb[(size_t)n16 * C_DIM + k0 + 1];
        b.x = W[n16 * HD_DIM + k0];        // B[d][e] = W[e][d]
        b.y = W[n16 * HD_DIM + k0 + 1];
        acc = wmma_k4(a, b, acc);
    }
#pragma unroll
    for (int r = 0; r < 8; ++r) {
        int m = r + half * 8;
        int token = tbase + m;
        int node = token / T_SEQ;
        int tt = token - node * T_SEQ;
        size_t dst;
        if (TP)
            dst = (((size_t)node * H_HEADS + h) * 16 + tt) * 16 + n16;
        else
            dst = (((size_t)tt * H_HEADS + h) * (size_t)nnodes + node) * 16 + n16;
        Xp[dst] = acc[r];
    }
}

// =====================================================================
// Spatial attention core (pre-Wo). One wave per (16 q-rows, t, h).
// Packed per-(t,h) layout: contiguous 1KB 16x16 fragment tiles.
// V tiles are double-buffered in LDS via async-to-LDS (overlapped with the
// energy WMMAs of the previous tile); next K tile is prefetched.
// Energies clipped to [-5,5] by the reference -> plain sum-of-exp softmax.
// MODE 0: m[q,k] = adj_sr[k,q] (leading dim R) / 1: adj_r[q,k] / 2: none
// =====================================================================
template <int MODE>
__global__ void k_sattn(const float* __restrict__ Qp, const float* __restrict__ Kp,
                        const float* __restrict__ Vp, const float* __restrict__ adj,
                        float* __restrict__ outc, int Nq, int Nk) {
    __shared__ float P[16 * 17];
    __shared__ float Vt[2][16 * 16];
    int lane = threadIdx.x, half = lane >> 4, n16 = lane & 15;
    int qbase = blockIdx.x * 16;
    int th = blockIdx.y;                  // t*H + h
    int t = th / H_HEADS;
    int h = th % H_HEADS;

    const float* Qb = Qp + ((size_t)th * Nq + qbase) * 16;
    v2f afrag[4];
#pragma unroll
    for (int c = 0; c < 4; ++c) {
        int k0 = c * 4 + half * 2;
        afrag[c].x = Qb[(size_t)n16 * 16 + k0];
        afrag[c].y = Qb[(size_t)n16 * 16 + k0 + 1];
    }

    const float* Kbase = Kp + (size_t)th * Nk * 16;
    const float* Vbase = Vp + (size_t)th * Nk * 16;
    unsigned ldsv[2];
    ldsv[0] = (unsigned)(size_t)(&Vt[0][0]) + (unsigned)lane * 16u;
    ldsv[1] = (unsigned)(size_t)(&Vt[1][0]) + (unsigned)lane * 16u;

    v8f o = {0,0,0,0,0,0,0,0};
    float rs[8];
#pragma unroll
    for (int r = 0; r < 8; ++r) rs[r] = 0.f;

    int nkt = Nk / 16;
    async_tile_1k(Vbase, ldsv[0], lane);                // prime tile 0
    for (int kt = 0; kt < nkt; ++kt) {
        int kbase = kt * 16;
        if (kt + 1 < nkt) {                              // stage tile kt+1
            async_tile_1k(Vbase + (size_t)(kbase + 16) * 16, ldsv[(kt + 1) & 1], lane);
            __builtin_prefetch(Kbase + (size_t)(kbase + 16) * 16, 0, 1);
        }

        // --- energy = Qh * Kh^T (K = HD = 16) ---
        const float* Kb = Kbase + (size_t)kbase * 16;
        v8f e = {0,0,0,0,0,0,0,0};
#pragma unroll
        for (int c = 0; c < 4; ++c) {
            int k0 = c * 4 + half * 2;
            v2f b;
            b.x = Kb[(size_t)n16 * 16 + k0];   // B[d][kcol] = kh[kcol][d]
            b.y = Kb[(size_t)n16 * 16 + k0 + 1];
            e = wmma_k4(afrag[c], b, e);
        }

        int kk = kbase + n16;
        float madj[8];
        if (MODE == 0) {                       // 8 consecutive floats -> 2x b128
            const float* arow = adj + (size_t)kk * R_REG + qbase + half * 8;
            float4 m0 = *(const float4*)arow;
            float4 m1 = *(const float4*)(arow + 4);
            madj[0] = m0.x; madj[1] = m0.y; madj[2] = m0.z; madj[3] = m0.w;
            madj[4] = m1.x; madj[5] = m1.y; madj[6] = m1.z; madj[7] = m1.w;
        }
#pragma unroll
        for (int r = 0; r < 8; ++r) {
            float ev = e[r];
            if (MODE == 0) {
                float m = madj[r];
                ev = (m == 0.f) ? 1.0e6f : ev * m;
            } else if (MODE == 1) {
                int qi = qbase + r + half * 8;
                float m = adj[(size_t)qi * Nk + kk];
                ev = (m == 0.f) ? 1.0e6f : ev * m;
            }
            ev = fminf(fmaxf(ev * 0.125f, -5.f), 5.f);   // /sqrt(64), clip
            float pv = __expf(ev);
            float s = pv;                                 // row-sum within 16-lane half
            s += __shfl_xor(s, 1); s += __shfl_xor(s, 2);
            s += __shfl_xor(s, 4); s += __shfl_xor(s, 8);
            rs[r] += s;
            P[(r + half * 8) * 17 + n16] = pv;            // D-layout -> LDS
        }
        __syncthreads();
        // in-order async completion: <=2 outstanding ==> tile kt is resident
        if (kt + 1 < nkt) asm volatile("s_wait_asynccnt 0x2" ::: "memory");
        else              asm volatile("s_wait_asynccnt 0x0" ::: "memory");

        // --- O += P * Vh (K = 16 k-cols), P + V from LDS ---
        const float* Vl = &Vt[kt & 1][0];
#pragma unroll
        for (int c = 0; c < 4; ++c) {
            int k0 = c * 4 + half * 2;
            v2f pa, bv;
            pa.x = P[n16 * 17 + k0];
            pa.y = P[n16 * 17 + k0 + 1];
            bv.x = Vl[k0 * 16 + n16];                     // B[k][d] = vh[k][d]
            bv.y = Vl[(k0 + 1) * 16 + n16];
            o = wmma_k4(pa, bv, o);
        }
        __syncthreads();
    }

    // output back in (node, t, c) layout for the Wo GEMM
    const size_t rstr = (size_t)T_SEQ * C_DIM;
    float* Ob = outc + (size_t)qbase * rstr + (size_t)t * C_DIM + h * HD_DIM;
#pragma unroll
    for (int r = 0; r < 8; ++r) {
        int m = r + half * 8;
        Ob[(size_t)m * rstr + n16] = o[r] / rs[r];
    }
}

// =====================================================================
// Generic Y = [relu](X @ W^T + b). One wave per (16 tokens, 16 out-cols).
// =====================================================================
__global__ void k_gemm_xwt(const float* __restrict__ X, const float* __restrict__ W,
                           const float* __restrict__ bias, float* __restrict__ Y,
                           int ntok, int cin, int cout, int relu) {
    int lane = threadIdx.x, half = lane >> 4, n16 = lane & 15;
    int tbase = blockIdx.x * 16;
    int obase = blockIdx.y * 16;
    if (tbase >= ntok) return;
    const float* Xb = X + (size_t)tbase * cin;
    const float* Wb = W + (size_t)obase * cin;
    v8f acc = {0,0,0,0,0,0,0,0};
    for (int c = 0; c < cin / 4; ++c) {
        int k0 = c * 4 + half * 2;
        v2f a, b;
        a.x = Xb[(size_t)n16 * cin + k0];
        a.y = Xb[(size_t)n16 * cin + k0 + 1];
        b.x = Wb[(size_t)n16 * cin + k0];    // B[d][o] = W[o][d]
        b.y = Wb[(size_t)n16 * cin + k0 + 1];
        acc = wmma_k4(a, b, acc);
    }
    float bo = bias[obase + n16];
#pragma unroll
    for (int r = 0; r < 8; ++r) {
        float v = acc[r] + bo;
        if (relu) v = fmaxf(v, 0.f);
        Y[(size_t)(tbase + r + half * 8) * cout + obase + n16] = v;
    }
}

// =====================================================================
// Y = LayerNorm(A + B) * g + beta  (per 64-wide token, eps=1e-5, ddof=0)
// =====================================================================
__global__ void k_ln_res(const float* __restrict__ A, const float* __restrict__ B,
                         const float* __restrict__ g, const float* __restrict__ be,
                         float* __restrict__ Y, int ntok) {
    int idx = blockIdx.x * blockDim.x + threadIdx.x;
    if (idx >= ntok) return;
    const float* a = A + (size_t)idx * C_DIM;
    const float* b = B + (size_t)idx * C_DIM;
    float v[C_DIM];
    float s = 0.f, s2 = 0.f;
    for (int c = 0; c < C_DIM; ++c) {
        float x = a[c] + b[c];
        v[c] = x; s += x; s2 += x * x;
    }
    float mean = s * (1.f / C_DIM);
    float var = s2 * (1.f / C_DIM) - mean * mean;
    float rstd = rsqrtf(var + 1e-5f);
    float* y = Y + (size_t)idx * C_DIM;
    for (int c = 0; c < C_DIM; ++c)
        y[c] = (v[c] - mean) * rstd * g[c] + be[c];
}

// =====================================================================
// qt = x1 + Temb[(288 + t + tt) % 288]
// =====================================================================
__global__ void k_time_emb(const float* __restrict__ x1, const float* __restrict__ emb,
                           const int* __restrict__ tptr, float* __restrict__ qt, int total) {
    int i = blockIdx.x * blockDim.x + threadIdx.x;
    if (i >= total) return;
    int c = i % C_DIM;
    int tt = (i / C_DIM) % T_SEQ;
    int ind = (TIME_NUMC + tptr[0] + tt) % TIME_NUMC;
    if (ind < 0) ind += TIME_NUMC;
    qt[i] = x1[i] + emb[(size_t)ind * C_DIM + c];
}

// =====================================================================
// Temporal attention core (pre-Wo). One wave per (n, h); packed per-(n,h)
// 16x16 tiles (T=12 valid rows, rest masked).
// =====================================================================
__global__ void k_tattn(const float* __restrict__ qh, const float* __restrict__ kh,
                        const float* __restrict__ vh, float* __restrict__ outc) {
    __shared__ float P[16 * 17];
    int lane = threadIdx.x, half = lane >> 4, n16 = lane & 15;
    int n = blockIdx.x / H_HEADS;
    int h = blockIdx.x % H_HEADS;
    size_t base = ((size_t)n * H_HEADS + h) * 256;   // 16x16 tile
    const float* Qb = qh + base;
    const float* Kb = kh + base;
    const float* Vb = vh + base;
    bool okrow = n16 < T_SEQ;
    v2f afrag[4];
#pragma unroll
    for (int c = 0; c < 4; ++c) {
        int k0 = c * 4 + half * 2;
        afrag[c].x = okrow ? Qb[(size_t)n16 * 16 + k0] : 0.f;
        afrag[c].y = okrow ? Qb[(size_t)n16 * 16 + k0 + 1] : 0.f;
    }
    v8f e = {0,0,0,0,0,0,0,0};
#pragma unroll
    for (int c = 0; c < 4; ++c) {
        int k0 = c * 4 + half * 2;
        v2f b;
        b.x = okrow ? Kb[(size_t)n16 * 16 + k0] : 0.f;
        b.y = okrow ? Kb[(size_t)n16 * 16 + k0 + 1] : 0.f;
        e = wmma_k4(afrag[c], b, e);
    }
    float rs[8];
#pragma unroll
    for (int r = 0; r < 8; ++r) {
        float ev = fminf(fmaxf(e[r] * 0.125f, -5.f), 5.f);
        float pv = okrow ? __expf(ev) : 0.f;  // kill padded k-columns
        float s = pv;
        s += __shfl_xor(s, 1); s += __shfl_xor(s, 2);
        s += __shfl_xor(s, 4); s += __shfl_xor(s, 8);
        rs[r] = s;
        P[(r + half * 8) * 17 + n16] = pv;
    }
    __syncthreads();
    v8f o = {0,0,0,0,0,0,0,0};
#pragma unroll
    for (int c = 0; c < 4; ++c) {
        int k0 = c * 4 + half * 2;
        v2f pa, bv;
        pa.x = P[n16 * 17 + k0];
        pa.y = P[n16 * 17 + k0 + 1];
        bv.x = (k0     < T_SEQ) ? Vb[(size_t)k0 * 16 + n16] : 0.f;
        bv.y = (k0 + 1 < T_SEQ) ? Vb[(size_t)(k0 + 1) * 16 + n16] : 0.f;
        o = wmma_k4(pa, bv, o);
    }
    float* Ob = outc + (size_t)n * T_SEQ * C_DIM + h * HD_DIM;
#pragma unroll
    for (int r = 0; r < 8; ++r) {
        int m = r + half * 8;
        if (m < T_SEQ) Ob[(size_t)m * C_DIM + n16] = o[r] / rs[r];
    }
}

// =====================================================================
// Host-side orchestration
// =====================================================================
static inline dim3 g1(size_t n, int b) { return dim3((unsigned)((n + b - 1) / b)); }

extern "C" void kernel_launch(void* const* d_in, const int* in_sizes, int n_in,
                              void* d_out, int out_size, void* d_ws, size_t ws_size,
                              hipStream_t stream) {
    (void)in_sizes; (void)n_in; (void)out_size; (void)ws_size;
    // inputs in setup_inputs() dict order; params flattened with sorted dict keys
    const float* value  = (const float*)d_in[0];
    const float* key_in = (const float*)d_in[1];
    const float* query  = (const float*)d_in[2];
    const float* adj_sr = (const float*)d_in[3];
    const float* adj_r  = (const float*)d_in[4];
    /* d_in[5] = d (unused) */
    const int*   t_ptr  = (const int*)d_in[6];
    const float* S_DS   = (const float*)d_in[7];
    const float* S_I    = (const float*)d_in[8];
    const float* S_embW = (const float*)d_in[9];
    const float* S_embb = (const float*)d_in[10];
    const float* S_ffW1 = (const float*)d_in[11];
    const float* S_ffW2 = (const float*)d_in[12];
    const float* S_ffb1 = (const float*)d_in[13];
    const float* S_ffb2 = (const float*)d_in[14];
    const float* S_fsW  = (const float*)d_in[15];
    const float* S_fsb  = (const float*)d_in[16];
    const float* S_ln1b = (const float*)d_in[17];
    const float* S_ln1g = (const float*)d_in[18];
    const float* S_ln2b = (const float*)d_in[19];
    const float* S_ln2g = (const float*)d_in[20];
    const float* rr_Wk  = (const float*)d_in[21];
    const float* rr_Wo  = (const float*)d_in[22];
    const float* rr_Wq  = (const float*)d_in[23];
    const float* rr_Wv  = (const float*)d_in[24];
    const float* rr_bo  = (const float*)d_in[25];
    const float* rs_Wk  = (const float*)d_in[26];
    const float* rs_Wo  = (const float*)d_in[27];
    const float* rs_Wq  = (const float*)d_in[28];
    const float* rs_Wv  = (const float*)d_in[29];
    const float* rs_bo  = (const float*)d_in[30];
    const float* sr_Wk  = (const float*)d_in[31];
    const float* sr_Wo  = (const float*)d_in[32];
    const float* sr_Wq  = (const float*)d_in[33];
    const float* sr_Wv  = (const float*)d_in[34];
    const float* sr_bo  = (const float*)d_in[35];
    const float* Ta_Wk  = (const float*)d_in[36];
    const float* Ta_Wo  = (const float*)d_in[37];
    const float* Ta_Wq  = (const float*)d_in[38];
    const float* Ta_Wv  = (const float*)d_in[39];
    const float* Ta_bo  = (const float*)d_in[40];
    const float* T_emb  = (const float*)d_in[41];
    const float* T_ffW1 = (const float*)d_in[42];
    const float* T_ffW2 = (const float*)d_in[43];
    const float* T_ffb1 = (const float*)d_in[44];
    const float* T_ffb2 = (const float*)d_in[45];
    const float* T_ln1b = (const float*)d_in[46];
    const float* T_ln1g = (const float*)d_in[47];
    const float* T_ln2b = (const float*)d_in[48];
    const float* T_ln2g = (const float*)d_in[49];
    const float* b_ln1b = (const float*)d_in[50];
    const float* b_ln1g = (const float*)d_in[51];
    const float* b_ln2b = (const float*)d_in[52];
    const float* b_ln2g = (const float*)d_in[53];

    const size_t tokN = (size_t)N_NODES * T_SEQ;   // 24576
    const size_t tokR = (size_t)R_REG * T_SEQ;     // 3072
    const size_t big = tokN * C_DIM;               // 1.57 M floats
    const size_t small = tokR * C_DIM;             // 196 K floats
    const size_t bigT = (size_t)N_NODES * H_HEADS * 256; // temporal packed tiles

    float* ws = (float*)d_ws;                      // ~82 MB total
    float* B0 = ws;            float* B1 = B0 + big;  float* B2 = B1 + big;
    float* B3 = B2 + big;      float* B4 = B3 + big;  float* B5 = B4 + big;
    float* B6 = B5 + big;      float* B7 = B6 + big;
    float* FF = B7 + big;                          // tokN*256 floats == 3*bigT
    float* R0 = FF + tokN * 256;
    float* R1 = R0 + small; float* R2 = R1 + small; float* R3 = R2 + small;
    float* R4 = R3 + small; float* R5 = R4 + small; float* R6 = R5 + small;
    float* R7 = R6 + small;
    float* TP0 = FF;           // temporal packed vht  (FF dead by then)
    float* TP1 = FF + bigT;    // temporal packed kht
    float* TP2 = FF + 2*bigT;  // temporal packed qht

    dim3 w32(32);
    const int totN = (int)big;

    // ---- spatial branch ----
    k_add_ds<<<g1(big, 256), 256, 0, stream>>>(query, key_in, S_DS, S_embW, S_embb, B0, B1, totN);
    k_proj_pack<0><<<dim3(tokN / 16, H_HEADS), w32, 0, stream>>>(value, sr_Wv, B3, (int)tokN, N_NODES);
    k_proj_pack<0><<<dim3(tokN / 16, H_HEADS), w32, 0, stream>>>(B1,    sr_Wk, B2, (int)tokN, N_NODES);
    k_proj_pack<0><<<dim3(tokR / 16, H_HEADS), w32, 0, stream>>>(S_I,   sr_Wq, R0, (int)tokR, R_REG);
    k_sattn<0><<<dim3(R_REG / 16, T_SEQ * H_HEADS), w32, 0, stream>>>(R0, B2, B3, adj_sr, R1, R_REG, N_NODES);
    k_gemm_xwt<<<dim3(tokR / 16, 4), w32, 0, stream>>>(R1, sr_Wo, sr_bo, R2, (int)tokR, 64, 64, 0);

    k_proj_pack<0><<<dim3(tokR / 16, H_HEADS), w32, 0, stream>>>(R2, rr_Wv, R3, (int)tokR, R_REG);
    k_proj_pack<0><<<dim3(tokR / 16, H_HEADS), w32, 0, stream>>>(R2, rr_Wk, R4, (int)tokR, R_REG);
    k_proj_pack<0><<<dim3(tokR / 16, H_HEADS), w32, 0, stream>>>(R2, rr_Wq, R5, (int)tokR, R_REG);
    k_sattn<1><<<dim3(R_REG / 16, T_SEQ * H_HEADS), w32, 0, stream>>>(R5, R4, R3, adj_r, R6, R_REG, R_REG);
    k_gemm_xwt<<<dim3(tokR / 16, 4), w32, 0, stream>>>(R6, rr_Wo, rr_bo, R7, (int)tokR, 64, 64, 0);

    k_proj_pack<0><<<dim3(tokR / 16, H_HEADS), w32, 0, stream>>>(R7, rs_Wv, R0, (int)tokR, R_REG);
    k_proj_pack<0><<<dim3(tokR / 16, H_HEADS), w32, 0, stream>>>(R7, rs_Wk, R1, (int)tokR, R_REG);
    k_proj_pack<0><<<dim3(tokN / 16, H_HEADS), w32, 0, stream>>>(B0, rs_Wq, B1, (int)tokN, N_NODES);
    k_sattn<2><<<dim3(N_NODES / 16, T_SEQ * H_HEADS), w32, 0, stream>>>(B1, R1, R0, nullptr, B2, N_NODES, R_REG);
    k_gemm_xwt<<<dim3(tokN / 16, 4), w32, 0, stream>>>(B2, rs_Wo, rs_bo, B3, (int)tokN, 64, 64, 0);

    k_ln_res<<<g1(tokN, 256), 256, 0, stream>>>(B3, B0, S_ln1g, S_ln1b, B4, (int)tokN);       // x
    k_gemm_xwt<<<dim3(tokN / 16, 16), w32, 0, stream>>>(B4, S_ffW1, S_ffb1, FF, (int)tokN, 64, 256, 1);
    k_gemm_xwt<<<dim3(tokN / 16, 4),  w32, 0, stream>>>(FF, S_ffW2, S_ffb2, B0, (int)tokN, 256, 64, 0);
    k_ln_res<<<g1(tokN, 256), 256, 0, stream>>>(B0, B4, S_ln2g, S_ln2b, B1, (int)tokN);       // U
    k_gemm_xwt<<<dim3(tokN / 16, 4), w32, 0, stream>>>(B1, S_fsW, S_fsb, B2, (int)tokN, 64, 64, 0);
    k_ln_res<<<g1(tokN, 256), 256, 0, stream>>>(B2, query, b_ln1g, b_ln1b, B3, (int)tokN);    // x1

    // ---- temporal branch ----
    k_time_emb<<<g1(big, 256), 256, 0, stream>>>(B3, T_emb, t_ptr, B0, totN);                 // qt
    k_proj_pack<1><<<dim3(tokN / 16, H_HEADS), w32, 0, stream>>>(B3, Ta_Wv, TP0, (int)tokN, N_NODES); // vht(x1)
    k_proj_pack<1><<<dim3(tokN / 16, H_HEADS), w32, 0, stream>>>(B0, Ta_Wk, TP1, (int)tokN, N_NODES); // kht(qt)
    k_proj_pack<1><<<dim3(tokN / 16, H_HEADS), w32, 0, stream>>>(B0, Ta_Wq, TP2, (int)tokN, N_NODES); // qht(qt)
    k_tattn<<<dim3(N_NODES * H_HEADS), w32, 0, stream>>>(TP2, TP1, TP0, B6);
    k_gemm_xwt<<<dim3(tokN / 16, 4), w32, 0, stream>>>(B6, Ta_Wo, Ta_bo, B7, (int)tokN, 64, 64, 0);
    k_ln_res<<<g1(tokN, 256), 256, 0, stream>>>(B7, B0, T_ln1g, T_ln1b, B4, (int)tokN);       // xt
    k_gemm_xwt<<<dim3(tokN / 16, 16), w32, 0, stream>>>(B4, T_ffW1, T_ffb1, FF, (int)tokN, 64, 256, 1);
    k_gemm_xwt<<<dim3(tokN / 16, 4),  w32, 0, stream>>>(FF, T_ffW2, T_ffb2, B0, (int)tokN, 256, 64, 0);
    k_ln_res<<<g1(tokN, 256), 256, 0, stream>>>(B0, B4, T_ln2g, T_ln2b, B2, (int)tokN);       // tout
    k_ln_res<<<g1(tokN, 256), 256, 0, stream>>>(B2, B3, b_ln2g, b_ln2b, (float*)d_out, (int)tokN);
}